// VGAE_70712341561936
// MI455X (gfx1250) — compile-verified
//
#include <hip/hip_runtime.h>
#include <math.h>

typedef float v2f __attribute__((ext_vector_type(2)));
typedef float v8f __attribute__((ext_vector_type(8)));

#define NN    8192
#define FF    512
#define HH1   32
#define HH2   16
#define EE    262144

// Workspace layout (float offsets). Total ~4.72 MB.
#define OFF_XW1   0                      // N*H1 = 262144
#define OFF_FEAT  (NN*HH1)               // 262144 (spmm accum, pre-relu)
#define OFF_T2    (2*NN*HH1)             // 524288
#define OFF_T3    (OFF_T2 + NN*HH2)      // 655360
#define OFF_MU    (OFF_T3 + NN*HH2)      // 786432
#define OFF_LS    (OFF_MU + NN*HH2)      // 917504  (contiguous after MU)
#define OFF_Z     (OFF_LS + NN*HH2)      // 1048576
#define OFF_ACC   (OFF_Z  + NN*HH2)      // 1179648 -> byte 4718592 (8B aligned)
// acc[0] = bce sum (double), acc[1] = kl sum (double)

#define LOG2E_F   1.4426950408889634f
#define LN2_F     0.6931471805599453f

// ---------------------------------------------------------------- init: zero accumulation targets
__global__ void vgae_init_kernel(float* wsf) {
    int idx = blockIdx.x * 256 + threadIdx.x;            // 262144 threads
    if (idx < NN*HH1) {
        wsf[OFF_FEAT + idx] = 0.0f;                      // feat accum
        wsf[OFF_MU   + idx] = 0.0f;                      // mu + ls accum (contiguous 262144)
    }
    if (idx == 0) {
        double* acc = (double*)(wsf + OFF_ACC);
        acc[0] = 0.0; acc[1] = 0.0;
    }
}

// ---------------------------------------------------------------- XW1 = features @ W1   [N,F]x[F,H1]
__global__ void vgae_gemm1_kernel(const float* __restrict__ feats,
                                  const float* __restrict__ W1,
                                  float* __restrict__ xw1) {
    int idx = blockIdx.x * 256 + threadIdx.x;            // N*H1 = 262144
    int row = idx >> 5, col = idx & 31;
    const float* fr = feats + row * FF;
    float acc = 0.0f;
    #pragma unroll 4
    for (int f = 0; f < FF; ++f)
        acc = fmaf(fr[f], W1[f*HH1 + col], acc);
    xw1[idx] = acc;
}

// ---------------------------------------------------------------- feat += spmm(XW1): one wave per edge
__global__ void vgae_spmm1_kernel(const int* __restrict__ rows,
                                  const int* __restrict__ cols,
                                  const float* __restrict__ vals,
                                  const float* __restrict__ xw1,
                                  float* __restrict__ feat) {
    int tid  = threadIdx.x;
    int lane = tid & 31;
    int e = blockIdx.x * 8 + (tid >> 5);                 // E/8 blocks
    int r = rows[e], c = cols[e];
    float v = vals[e];
    atomicAdd(&feat[r*HH1 + lane], v * xw1[c*HH1 + lane]);
}

// ---------------------------------------------------------------- t2 = relu(feat)@W2, t3 = relu(feat)@W3
__global__ void vgae_gemm23_kernel(const float* __restrict__ feat,
                                   const float* __restrict__ W2,
                                   const float* __restrict__ W3,
                                   float* __restrict__ t2,
                                   float* __restrict__ t3) {
    __shared__ float w2s[HH1*HH2];
    __shared__ float w3s[HH1*HH2];
    int tid = threadIdx.x;
    for (int i = tid; i < HH1*HH2; i += 256) { w2s[i] = W2[i]; w3s[i] = W3[i]; }
    __syncthreads();
    int idx = blockIdx.x * 256 + tid;                    // N*H2 = 131072
    int row = idx >> 4, col = idx & 15;
    const float* fr = feat + row * HH1;
    float a2 = 0.0f, a3 = 0.0f;
    #pragma unroll
    for (int k = 0; k < HH1; ++k) {
        float v = fmaxf(fr[k], 0.0f);
        a2 = fmaf(v, w2s[k*HH2 + col], a2);
        a3 = fmaf(v, w3s[k*HH2 + col], a3);
    }
    t2[idx] = a2; t3[idx] = a3;
}

// ---------------------------------------------------------------- mu += spmm(t2), ls += spmm(t3)
__global__ void vgae_spmm2_kernel(const int* __restrict__ rows,
                                  const int* __restrict__ cols,
                                  const float* __restrict__ vals,
                                  const float* __restrict__ t2,
                                  const float* __restrict__ t3,
                                  float* __restrict__ mu,
                                  float* __restrict__ ls) {
    int tid  = threadIdx.x;
    int lane = tid & 31;
    int e = blockIdx.x * 8 + (tid >> 5);
    int r = rows[e], c = cols[e];
    float v = vals[e];
    if (lane < HH2) {
        atomicAdd(&mu[r*HH2 + lane], v * t2[c*HH2 + lane]);
    } else {
        int l = lane - HH2;
        atomicAdd(&ls[r*HH2 + l], v * t3[c*HH2 + l]);
    }
}

// ---------------------------------------------------------------- z = mu + eps*exp(ls); KL partial sum
__global__ void vgae_zkl_kernel(const float* __restrict__ mu,
                                const float* __restrict__ ls,
                                const float* __restrict__ eps,
                                float* __restrict__ z,
                                double* __restrict__ acc) {
    __shared__ float red[256];
    int tid = threadIdx.x;
    int idx = blockIdx.x * 256 + tid;                    // N*H2 = 131072
    float m = mu[idx], s = ls[idx];
    z[idx] = m + eps[idx] * expf(s);
    red[tid] = 1.0f + 2.0f*s - m*m - expf(2.0f*s);
    __syncthreads();
    for (int st = 128; st > 0; st >>= 1) {
        if (tid < st) red[tid] += red[tid + st];
        __syncthreads();
    }
    if (tid == 0) atomicAdd(&acc[1], (double)red[0]);
}

// ---------------------------------------------------------------- fused decode: WMMA z@z^T tile + BCE reduce
// One wave32 per 16x16 logits tile; K=16 via 4x V_WMMA_F32_16X16X4_F32 (exact f32).
// BCE uses raw v_exp_f32 / v_log_f32 (base-2 HW transcendentals):
//   softplus(-x) = max(-x,0) + ln2 * log2(1 + exp2(-log2e*|x|))
__global__ void vgae_decode_bce_kernel(const float* __restrict__ z,
                                       const float* __restrict__ adj_label,
                                       const float* __restrict__ pw_p,
                                       double* __restrict__ acc) {
    const float pwm1 = pw_p[0] - 1.0f;
    int tid  = threadIdx.x;
    int lane = tid & 31;
    int lm   = lane & 15;        // M/N index within tile
    int lh   = lane >> 4;        // lane-half selects K pair / row-half
    long long waveId = (long long)blockIdx.x * 8 + (tid >> 5);
    int jt = (int)(waveId & 511);                        // 512 j-tiles
    int it = (int)(waveId >> 9);                         // 512 i-tiles

    const float* zi = z + (it*16 + lm) * HH2;            // A: row lm of i-tile
    const float* zj = z + (jt*16 + lm) * HH2;            // B (=z^T col): row lm of j-tile

    v8f c = {0.f, 0.f, 0.f, 0.f, 0.f, 0.f, 0.f, 0.f};
    #pragma unroll
    for (int ck = 0; ck < 4; ++ck) {
        int off = ck*4 + lh*2;                           // ISA 16x4-f32 A/B lane layout
        v2f a = *(const v2f*)(zi + off);
        v2f b = *(const v2f*)(zj + off);
        c = __builtin_amdgcn_wmma_f32_16x16x4_f32(false, a, false, b, (short)0, c, false, false);
    }

    // C/D layout: VGPR r, lane -> M = r + 8*lh, N = lm
    float lsum = 0.0f;
    int jg = jt*16 + lm;
    const float* lrow = adj_label + (long long)(it*16 + 8*lh) * NN + jg;
    #pragma unroll
    for (int r = 0; r < 8; ++r) {
        float x = c[r];
        // stream labels once; NT hint keeps z resident in L2
        float label = __builtin_nontemporal_load(lrow + (long long)r * NN);
        float e  = __builtin_amdgcn_exp2f(-LOG2E_F * fabsf(x));     // exp(-|x|)
        float sp = fmaxf(-x, 0.0f) + LN2_F * __builtin_amdgcn_logf(1.0f + e);
        float coef = fmaf(pwm1, label, 1.0f);                        // 1 + (pw-1)*label
        lsum += fmaf(-label, x, x);                                  // (1-label)*x
        lsum  = fmaf(coef, sp, lsum);
    }

    __shared__ float red[256];
    red[tid] = lsum;
    __syncthreads();
    for (int st = 128; st > 0; st >>= 1) {
        if (tid < st) red[tid] += red[tid + st];
        __syncthreads();
    }
    if (tid == 0) atomicAdd(&acc[0], (double)red[0]);
}

// ---------------------------------------------------------------- combine
__global__ void vgae_final_kernel(const double* __restrict__ acc,
                                  const float* __restrict__ norm_p,
                                  float* __restrict__ out) {
    double invNN = 1.0 / ((double)NN * (double)NN);
    double loss = (double)norm_p[0] * acc[0] * invNN - 0.5 * acc[1] * invNN;
    out[0] = (float)loss;
}

extern "C" void kernel_launch(void* const* d_in, const int* in_sizes, int n_in,
                              void* d_out, int out_size, void* d_ws, size_t ws_size,
                              hipStream_t stream) {
    const float* features  = (const float*)d_in[0];
    const int*   adj_rows  = (const int*)  d_in[1];
    const int*   adj_cols  = (const int*)  d_in[2];
    const float* adj_vals  = (const float*)d_in[3];
    const float* adj_label = (const float*)d_in[4];
    const float* eps       = (const float*)d_in[5];
    const float* W1        = (const float*)d_in[6];
    const float* W2        = (const float*)d_in[7];
    const float* W3        = (const float*)d_in[8];
    const float* norm_p    = (const float*)d_in[9];
    const float* pw_p      = (const float*)d_in[10];

    float*  wsf = (float*)d_ws;
    float*  xw1  = wsf + OFF_XW1;
    float*  feat = wsf + OFF_FEAT;
    float*  t2   = wsf + OFF_T2;
    float*  t3   = wsf + OFF_T3;
    float*  mu   = wsf + OFF_MU;
    float*  ls   = wsf + OFF_LS;
    float*  z    = wsf + OFF_Z;
    double* acc  = (double*)(wsf + OFF_ACC);
    float*  out  = (float*)d_out;

    vgae_init_kernel  <<<NN*HH1/256, 256, 0, stream>>>(wsf);
    vgae_gemm1_kernel <<<NN*HH1/256, 256, 0, stream>>>(features, W1, xw1);
    vgae_spmm1_kernel <<<EE/8,       256, 0, stream>>>(adj_rows, adj_cols, adj_vals, xw1, feat);
    vgae_gemm23_kernel<<<NN*HH2/256, 256, 0, stream>>>(feat, W2, W3, t2, t3);
    vgae_spmm2_kernel <<<EE/8,       256, 0, stream>>>(adj_rows, adj_cols, adj_vals, t2, t3, mu, ls);
    vgae_zkl_kernel   <<<NN*HH2/256, 256, 0, stream>>>(mu, ls, eps, z, acc);
    // 512x512 tiles, 1 wave each, 8 waves per block -> 32768 blocks
    vgae_decode_bce_kernel<<<(512*512)/8, 256, 0, stream>>>(z, adj_label, pw_p, acc);
    vgae_final_kernel <<<1, 1, 0, stream>>>(acc, norm_p, out);
}